// EdgePredModel_8710193677021
// MI455X (gfx1250) — compile-verified
//
#include <hip/hip_runtime.h>

// CDNA5 / gfx1250, wave32. fp32-exact matrix math via V_WMMA_F32_16X16X4_F32.

typedef __attribute__((ext_vector_type(2))) float v2f;
typedef __attribute__((ext_vector_type(8))) float v8f;

#define NNODES 50000
#define FDIM   256     // F_IN == F_OUT == H == 256
#define KCAT   512     // concat K-dim (processed as 2 phases of 256)
#define MTILE  32      // rows (nodes or edges) per block
// block = 256 threads = 8 waves: wave w -> m-tile (w>>2), column group (w&3)
// (64 columns = four 16x16 accumulator tiles per wave).

static __device__ __forceinline__ v8f wmma_f32(v2f a, v2f b, v8f c) {
    return __builtin_amdgcn_wmma_f32_16x16x4_f32(
        false, a, false, b, (short)0, c, false, false);
}

// A tile LDS layout: addr = (k>>2)*(MTILE*4) + m*4 + (k&3)
// -> one aligned ds_load_b64 per lane per k-step, 32 distinct banks.
#define A_IDX(k, m) (((k) >> 2) * (MTILE * 4) + (m) * 4 + ((k) & 3))

// ---------------------------------------------------------------- zero ws
__global__ void k_zero(float* __restrict__ p, int n) {
    int i = blockIdx.x * blockDim.x + threadIdx.x;
    int stride = gridDim.x * blockDim.x;
    for (; i < n; i += stride) p[i] = 0.0f;
}

// ------------------------------------------------------- scatter mean prep
__global__ void k_scatter(const float* __restrict__ x,
                          const int* __restrict__ src,
                          const int* __restrict__ dst,
                          float* __restrict__ agg,
                          float* __restrict__ deg, int E) {
    int e = blockIdx.x;
    if (e >= E) return;
    int s = src[e];
    int d = dst[e];
    int f = threadIdx.x;
    atomicAdd(&agg[d * FDIM + f], x[s * FDIM + f]);
    if (f == 0) atomicAdd(&deg[d], 1.0f);
}

// --------------------------------------------------------- node-level GEMM
// h = relu( (agg/deg) @ W_l^T + x @ W_r^T + b_l )
__global__ __launch_bounds__(256) void k_node(
    const float* __restrict__ x, const float* __restrict__ agg,
    const float* __restrict__ deg, const float* __restrict__ Wl,
    const float* __restrict__ bl, const float* __restrict__ Wr,
    float* __restrict__ h) {
    __shared__ float lds_a[(FDIM / 4) * (MTILE * 4)];   // 32 KB

    const int tid  = threadIdx.x;
    const int base = blockIdx.x * MTILE;
    const int lane = tid & 31;
    const int half = lane >> 4;
    const int lm   = lane & 15;
    const int wave = tid >> 5;
    const int mt   = wave >> 2;          // 0..1  (m-tile)
    const int n0   = (wave & 3) * 64 + lm;
    const int mrow = mt * 16 + lm;       // this lane's A row

    v8f acc0 = {}, acc1 = {}, acc2 = {}, acc3 = {};

    for (int p = 0; p < 2; ++p) {
        // phase 0: A = agg/deg (weights W_l); phase 1: A = x (weights W_r)
        for (int idx = tid; idx < MTILE * FDIM; idx += 256) {
            int m = idx >> 8;            // node-in-tile (consecutive f per tid)
            int f = idx & 255;
            int v = base + m;
            float val = 0.0f;
            if (v < NNODES) {
                val = (p == 0)
                          ? agg[v * FDIM + f] * (1.0f / fmaxf(deg[v], 1.0f))
                          : x[v * FDIM + f];
            }
            lds_a[A_IDX(f, m)] = val;
        }
        __syncthreads();

        const float* W = (p == 0) ? Wl : Wr;
        for (int kk = 0; kk < FDIM; kk += 4) {
            float2 av = *(const float2*)&lds_a[A_IDX(kk + 2 * half, mrow)];
            v2f a; a.x = av.x; a.y = av.y;
            int ka = kk + 2 * half;
            float2 w0 = *(const float2*)&W[(n0)      * FDIM + ka];
            float2 w1 = *(const float2*)&W[(n0 + 16) * FDIM + ka];
            float2 w2 = *(const float2*)&W[(n0 + 32) * FDIM + ka];
            float2 w3 = *(const float2*)&W[(n0 + 48) * FDIM + ka];
            v2f b;
            b.x = w0.x; b.y = w0.y; acc0 = wmma_f32(a, b, acc0);
            b.x = w1.x; b.y = w1.y; acc1 = wmma_f32(a, b, acc1);
            b.x = w2.x; b.y = w2.y; acc2 = wmma_f32(a, b, acc2);
            b.x = w3.x; b.y = w3.y; acc3 = wmma_f32(a, b, acc3);
        }
        __syncthreads();   // LDS reused by next phase
    }

    // Epilogue: C tile — VGPR r, lanes 0-15: row r; lanes 16-31: row r+8
    float bb0 = bl[n0], bb1 = bl[n0 + 16], bb2 = bl[n0 + 32], bb3 = bl[n0 + 48];
    const int vbase = base + mt * 16 + 8 * half;
    if (base + MTILE <= NNODES) {
#pragma unroll
        for (int r = 0; r < 8; ++r) {
            float* row = &h[(vbase + r) * FDIM];
            row[n0]      = fmaxf(acc0[r] + bb0, 0.0f);
            row[n0 + 16] = fmaxf(acc1[r] + bb1, 0.0f);
            row[n0 + 32] = fmaxf(acc2[r] + bb2, 0.0f);
            row[n0 + 48] = fmaxf(acc3[r] + bb3, 0.0f);
        }
    } else {
#pragma unroll
        for (int r = 0; r < 8; ++r) {
            int v = vbase + r;
            if (v < NNODES) {
                float* row = &h[v * FDIM];
                row[n0]      = fmaxf(acc0[r] + bb0, 0.0f);
                row[n0 + 16] = fmaxf(acc1[r] + bb1, 0.0f);
                row[n0 + 32] = fmaxf(acc2[r] + bb2, 0.0f);
                row[n0 + 48] = fmaxf(acc3[r] + bb3, 0.0f);
            }
        }
    }
}

// ---------------------------------------------------- edge MLP (dominant)
// out[e] = W2 . relu( W1 @ concat(h[src], h[dst]) + b1 ) + b2
__global__ __launch_bounds__(256) void k_edge(
    const float* __restrict__ h, const int* __restrict__ src,
    const int* __restrict__ dst, const float* __restrict__ W1,
    const float* __restrict__ b1, const float* __restrict__ W2,
    const float* __restrict__ b2, float* __restrict__ out, int E) {
    __shared__ float lds_a[(FDIM / 4) * (MTILE * 4)];   // 32 KB
    __shared__ float edge_acc[MTILE];

    const int tid  = threadIdx.x;
    const int base = blockIdx.x * MTILE;
    if (tid < MTILE) edge_acc[tid] = 0.0f;

    const int lane = tid & 31;
    const int half = lane >> 4;
    const int lm   = lane & 15;
    const int wave = tid >> 5;
    const int mt   = wave >> 2;
    const int n0   = (wave & 3) * 64 + lm;
    const int mrow = mt * 16 + lm;

    v8f acc0 = {}, acc1 = {}, acc2 = {}, acc3 = {};

    for (int p = 0; p < 2; ++p) {
        // phase 0: gather h[src] (K columns 0..255 of W1)
        // phase 1: gather h[dst] (K columns 256..511 of W1)
        const int* idxp = (p == 0) ? src : dst;
        for (int idx = tid; idx < MTILE * FDIM; idx += 256) {
            int m = idx >> 8;
            int f = idx & 255;
            int e = base + m;
            float val = 0.0f;
            if (e < E) {
                int node = idxp[e];
                val = h[node * FDIM + f];
            }
            lds_a[A_IDX(f, m)] = val;
        }
        __syncthreads();

        const float* Wp = W1 + p * FDIM;   // column offset into K=512
        for (int kk = 0; kk < FDIM; kk += 4) {
            float2 av = *(const float2*)&lds_a[A_IDX(kk + 2 * half, mrow)];
            v2f a; a.x = av.x; a.y = av.y;
            int ka = kk + 2 * half;
            float2 w0 = *(const float2*)&Wp[(n0)      * KCAT + ka];
            float2 w1 = *(const float2*)&Wp[(n0 + 16) * KCAT + ka];
            float2 w2 = *(const float2*)&Wp[(n0 + 32) * KCAT + ka];
            float2 w3 = *(const float2*)&Wp[(n0 + 48) * KCAT + ka];
            v2f b;
            b.x = w0.x; b.y = w0.y; acc0 = wmma_f32(a, b, acc0);
            b.x = w1.x; b.y = w1.y; acc1 = wmma_f32(a, b, acc1);
            b.x = w2.x; b.y = w2.y; acc2 = wmma_f32(a, b, acc2);
            b.x = w3.x; b.y = w3.y; acc3 = wmma_f32(a, b, acc3);
        }
        __syncthreads();
    }

    // Fused bias + relu + dot(W2), then cross-lane / cross-wave reduction.
    float bb0 = b1[n0],      w20 = W2[n0];
    float bb1 = b1[n0 + 16], w21 = W2[n0 + 16];
    float bb2 = b1[n0 + 32], w22 = W2[n0 + 32];
    float bb3 = b1[n0 + 48], w23 = W2[n0 + 48];
    float pr[8];
#pragma unroll
    for (int r = 0; r < 8; ++r) {
        pr[r] = fmaxf(acc0[r] + bb0, 0.0f) * w20 +
                fmaxf(acc1[r] + bb1, 0.0f) * w21 +
                fmaxf(acc2[r] + bb2, 0.0f) * w22 +
                fmaxf(acc3[r] + bb3, 0.0f) * w23;
    }
#pragma unroll
    for (int r = 0; r < 8; ++r) {
#pragma unroll
        for (int off = 1; off < 16; off <<= 1) {
            pr[r] += __shfl_xor(pr[r], off, 32);
        }
    }
    if (lm == 0) {
#pragma unroll
        for (int r = 0; r < 8; ++r) {
            atomicAdd(&edge_acc[mt * 16 + r + 8 * half], pr[r]);
        }
    }
    __syncthreads();
    if (tid < MTILE) {
        int e = base + tid;
        if (e < E) out[e] = edge_acc[tid] + b2[0];
    }
}

// ------------------------------------------------------------------ launch
extern "C" void kernel_launch(void* const* d_in, const int* in_sizes, int n_in,
                              void* d_out, int out_size, void* d_ws, size_t ws_size,
                              hipStream_t stream) {
    const float* x  = (const float*)d_in[0];
    const int*   ei = (const int*)d_in[1];     // [2, E]
    const float* Wl = (const float*)d_in[2];
    const float* bl = (const float*)d_in[3];
    const float* Wr = (const float*)d_in[4];
    const float* W1 = (const float*)d_in[5];
    const float* b1 = (const float*)d_in[6];
    const float* W2 = (const float*)d_in[7];
    const float* b2 = (const float*)d_in[8];
    float* out = (float*)d_out;

    const int E = in_sizes[1] / 2;
    const int* src = ei;
    const int* dst = ei + E;

    // workspace: h [N*F] | agg [N*F] | deg [N]   (~102.6 MB fp32)
    float* h   = (float*)d_ws;
    float* agg = h + (size_t)NNODES * FDIM;
    float* deg = agg + (size_t)NNODES * FDIM;

    const int zcount = NNODES * FDIM + NNODES;   // agg and deg contiguous
    k_zero<<<2048, 256, 0, stream>>>(agg, zcount);
    k_scatter<<<E, 256, 0, stream>>>(x, src, dst, agg, deg, E);
    k_node<<<(NNODES + MTILE - 1) / MTILE, 256, 0, stream>>>(
        x, agg, deg, Wl, bl, Wr, h);
    k_edge<<<(E + MTILE - 1) / MTILE, 256, 0, stream>>>(
        h, src, dst, W1, b1, W2, b2, out, E);
}